// Adaptive_Pooling_Layer_71966472011817
// MI455X (gfx1250) — compile-verified
//
#include <hip/hip_runtime.h>
#include <hip/hip_bf16.h>
#include <math.h>

// Problem constants (from reference)
#define BB 16
#define NNODE 4000
#define DD 128
#define HH 4
#define KK 32
#define HK 128      // H*K
#define H1 16
#define H2 64

typedef __attribute__((ext_vector_type(2))) float v2f;
typedef __attribute__((ext_vector_type(8))) float v8f;

__device__ __forceinline__ v8f wmma4(v2f a, v2f b, v8f c) {
  // V_WMMA_F32_16X16X4_F32 : D = A(16x4) x B(4x16) + C(16x16)
  return __builtin_amdgcn_wmma_f32_16x16x4_f32(
      /*neg_a=*/false, a, /*neg_b=*/false, b,
      /*c_mod=*/(short)0, c, /*reuse_a=*/false, /*reuse_b=*/false);
}

// ---------------------------------------------------------------------------
// Kernel 1: fused mean-over-nodes + 3-layer centroid MLP.
// One thread per (b, d). Writes cent[b][hk][d] (hk = o = output feature).
// ---------------------------------------------------------------------------
__global__ void k_centroid_mlp(const float* __restrict__ node,
                               const float* __restrict__ w1, const float* __restrict__ b1,
                               const float* __restrict__ W2, const float* __restrict__ b2,
                               const float* __restrict__ W3, const float* __restrict__ b3,
                               float* __restrict__ cent) {
  int tid = blockIdx.x * blockDim.x + threadIdx.x;
  if (tid >= BB * DD) return;
  int b = tid / DD, d = tid % DD;

  const float* np = node + (size_t)b * NNODE * DD + d;
  float s = 0.f;
  for (int n = 0; n < NNODE; ++n) s += np[(size_t)n * DD];
  float m = s * (1.0f / (float)NNODE);

  float c1[H1];
#pragma unroll
  for (int i = 0; i < H1; ++i) c1[i] = fmaxf(m * w1[i] + b1[i], 0.f);

  float c2[H2];
  for (int j = 0; j < H2; ++j) {
    float a = b2[j];
#pragma unroll
    for (int i = 0; i < H1; ++i) a += W2[j * H1 + i] * c1[i];
    c2[j] = fmaxf(a, 0.f);
  }

  float* cp = cent + (size_t)b * HK * DD + d;
  for (int o = 0; o < HK; ++o) {
    float a = b3[o];
#pragma unroll 8
    for (int j = 0; j < H2; ++j) a += W3[o * H2 + j] * c2[j];
    cp[(size_t)o * DD] = fmaxf(a, 0.f);
  }
}

// ---------------------------------------------------------------------------
// Kernel 2: L2 norm of rows of length 128. Wave per row, shuffle reduce.
// ---------------------------------------------------------------------------
__global__ void k_rownorm128(const float* __restrict__ src, float* __restrict__ dst,
                             int rows) {
  int wid = (blockIdx.x * blockDim.x + threadIdx.x) >> 5;
  int lane = threadIdx.x & 31;
  if (wid >= rows) return;
  const float* r = src + (size_t)wid * 128;
  float s = 0.f;
#pragma unroll
  for (int i = 0; i < 4; ++i) {
    float v = r[lane + 32 * i];
    s += v * v;
  }
#pragma unroll
  for (int off = 16; off > 0; off >>= 1) s += __shfl_down(s, off, 32);
  if (lane == 0) dst[wid] = sqrtf(s);
}

// ---------------------------------------------------------------------------
// Kernel 3: dotsT[b][n][hk] = sum_d node[b,n,d] * cent[b,hk,d]   (WMMA f32)
// Wave computes a 16(n) x 128(hk) strip: 8 accumulators, K-loop over d.
// A(16x4) = node rows, B(4x16) = cent^T (cached).
// ---------------------------------------------------------------------------
__global__ void __launch_bounds__(256) k_dots_wmma(const float* __restrict__ node,
                                                   const float* __restrict__ cent,
                                                   float* __restrict__ dotsT) {
  int wid = (blockIdx.x * blockDim.x + threadIdx.x) >> 5;
  int lane = threadIdx.x & 31;
  if (wid >= BB * (NNODE / 16)) return;
  int b = wid / (NNODE / 16);
  int n0 = (wid % (NNODE / 16)) * 16;
  int half = lane >> 4;   // 0: K=0,1   1: K=2,3
  int l = lane & 15;

  const float* arow = node + ((size_t)b * NNODE + n0 + l) * DD;
  const float* centb = cent + (size_t)b * HK * DD;

  v8f acc[8];
  v8f zero = {0.f, 0.f, 0.f, 0.f, 0.f, 0.f, 0.f, 0.f};
#pragma unroll
  for (int t = 0; t < 8; ++t) acc[t] = zero;

  for (int d = 0; d < DD; d += 4) {
    int kk = d + half * 2;
    v2f a;
    a.x = arow[kk];
    a.y = arow[kk + 1];
#pragma unroll
    for (int t = 0; t < 8; ++t) {
      const float* bc = centb + (size_t)(t * 16 + l) * DD + kk;
      v2f bv;
      bv.x = bc[0];
      bv.y = bc[1];
      acc[t] = wmma4(a, bv, acc[t]);
    }
  }
  // C/D layout: VGPR v holds row M=v (lanes 0-15) / M=v+8 (lanes 16-31), col N=lane&15
#pragma unroll
  for (int t = 0; t < 8; ++t)
#pragma unroll
    for (int v = 0; v < 8; ++v)
      dotsT[((size_t)b * NNODE + (n0 + v + 8 * half)) * HK + t * 16 + l] = acc[t][v];
}

// ---------------------------------------------------------------------------
// Kernel 4: cosine normalize + per-head K-sum + 1x1 head-mix conv -> C[b][k][n]
// One thread per (b, n): stores coalesced across n for each k row.
// ---------------------------------------------------------------------------
__global__ void k_assign(const float* __restrict__ dotsT, const float* __restrict__ cn,
                         const float* __restrict__ nn_, const float* __restrict__ convw,
                         const float* __restrict__ convb, float* __restrict__ Cmat) {
  int tid = blockIdx.x * blockDim.x + threadIdx.x;
  if (tid >= BB * NNODE) return;
  int b = tid / NNODE, n = tid % NNODE;

  const float* drow = dotsT + (size_t)tid * HK;
  const float* cnb = cn + b * HK;
  float nv = nn_[tid];

  float s[HH] = {0.f, 0.f, 0.f, 0.f};
  for (int hk = 0; hk < HK; ++hk) {
    float cosv = drow[hk] / fmaxf(cnb[hk] * nv, 1e-6f);
    s[hk >> 5] += cosv;
  }
  float wmix[HH];
#pragma unroll
  for (int h = 0; h < HH; ++h) wmix[h] = convw[h] / (s[h] + 1e-10f);
  float cb = convb[0];

  for (int k = 0; k < KK; ++k) {
    float a = cb;
#pragma unroll
    for (int h = 0; h < HH; ++h) {
      int hk = h * KK + k;
      a += wmix[h] * (drow[hk] / fmaxf(cnb[hk] * nv, 1e-6f));
    }
    Cmat[((size_t)b * KK + k) * NNODE + n] = a;
  }
}

// ---------------------------------------------------------------------------
// Kernel 5: T = C @ adj    (the 1 GB streaming GEMM; adj read exactly once)
// Wave computes 32(k) x 16(m) tile, reduction over n in steps of 4 (WMMA f32).
// ---------------------------------------------------------------------------
__global__ void __launch_bounds__(256) k_cadj_wmma(const float* __restrict__ Cmat,
                                                   const float* __restrict__ adj,
                                                   float* __restrict__ T) {
  int wid = (blockIdx.x * blockDim.x + threadIdx.x) >> 5;
  int lane = threadIdx.x & 31;
  if (wid >= BB * (NNODE / 16)) return;
  int b = wid / (NNODE / 16);
  int m0 = (wid % (NNODE / 16)) * 16;
  int half = lane >> 4;
  int l = lane & 15;

  const float* Cb = Cmat + (size_t)b * KK * NNODE;
  const float* adjb = adj + (size_t)b * NNODE * NNODE;

  v8f zero = {0.f, 0.f, 0.f, 0.f, 0.f, 0.f, 0.f, 0.f};
  v8f acc0 = zero, acc1 = zero;

  for (int n = 0; n < NNODE; n += 4) {
    int kk = n + half * 2;
    // prefetch adj rows a few K-steps ahead (streams 1 GB through L2)
    __builtin_prefetch(adjb + (size_t)(kk + 16) * NNODE + m0 + l, 0, 1);
    v2f bv;
    bv.x = adjb[(size_t)kk * NNODE + m0 + l];
    bv.y = adjb[(size_t)(kk + 1) * NNODE + m0 + l];
    v2f a0, a1;
    a0.x = Cb[(size_t)l * NNODE + kk];
    a0.y = Cb[(size_t)l * NNODE + kk + 1];
    a1.x = Cb[(size_t)(l + 16) * NNODE + kk];
    a1.y = Cb[(size_t)(l + 16) * NNODE + kk + 1];
    acc0 = wmma4(a0, bv, acc0);
    acc1 = wmma4(a1, bv, acc1);
  }
#pragma unroll
  for (int v = 0; v < 8; ++v) {
    T[((size_t)b * KK + (v + 8 * half)) * NNODE + m0 + l] = acc0[v];
    T[((size_t)b * KK + (16 + v + 8 * half)) * NNODE + m0 + l] = acc1[v];
  }
}

// ---------------------------------------------------------------------------
// Kernel 6: P_partial[chunk][b][k][d] = sum_{n in chunk} C[b,k,n]*node[b,n,d]
// Wave computes 32(k) x 16(d) tile over a 1000-long n chunk (WMMA f32).
// ---------------------------------------------------------------------------
__global__ void __launch_bounds__(256) k_cnode_wmma(const float* __restrict__ Cmat,
                                                    const float* __restrict__ node,
                                                    float* __restrict__ Ppart) {
  int wid = (blockIdx.x * blockDim.x + threadIdx.x) >> 5;
  int lane = threadIdx.x & 31;
  if (wid >= BB * 8 * 4) return;
  int b = wid >> 5;
  int r = wid & 31;
  int dt = r >> 2;        // 8 d-tiles of 16
  int chunk = r & 3;      // 4 n-chunks of 1000
  int d0 = dt * 16;
  int nbeg = chunk * (NNODE / 4);
  int nend = nbeg + (NNODE / 4);
  int half = lane >> 4;
  int l = lane & 15;

  const float* Cb = Cmat + (size_t)b * KK * NNODE;
  const float* nodeb = node + (size_t)b * NNODE * DD;

  v8f zero = {0.f, 0.f, 0.f, 0.f, 0.f, 0.f, 0.f, 0.f};
  v8f acc0 = zero, acc1 = zero;

  for (int n = nbeg; n < nend; n += 4) {
    int kk = n + half * 2;
    v2f bv;
    bv.x = nodeb[(size_t)kk * DD + d0 + l];
    bv.y = nodeb[(size_t)(kk + 1) * DD + d0 + l];
    v2f a0, a1;
    a0.x = Cb[(size_t)l * NNODE + kk];
    a0.y = Cb[(size_t)l * NNODE + kk + 1];
    a1.x = Cb[(size_t)(l + 16) * NNODE + kk];
    a1.y = Cb[(size_t)(l + 16) * NNODE + kk + 1];
    acc0 = wmma4(a0, bv, acc0);
    acc1 = wmma4(a1, bv, acc1);
  }
  float* pp = Ppart + (size_t)chunk * BB * KK * DD + (size_t)b * KK * DD;
#pragma unroll
  for (int v = 0; v < 8; ++v) {
    pp[(size_t)(v + 8 * half) * DD + d0 + l] = acc0[v];
    pp[(size_t)(16 + v + 8 * half) * DD + d0 + l] = acc1[v];
  }
}

// ---------------------------------------------------------------------------
// Kernel 7: new_adj[b,k,j] = relu( sum_m T[b,k,m] * C[b,j,m] )
// Wave per (b,k,j), coalesced strided dot + shuffle reduce.
// ---------------------------------------------------------------------------
__global__ void k_newadj(const float* __restrict__ T, const float* __restrict__ Cmat,
                         float* __restrict__ out_adj) {
  int wid = (blockIdx.x * blockDim.x + threadIdx.x) >> 5;
  int lane = threadIdx.x & 31;
  if (wid >= BB * KK * KK) return;
  int b = wid / (KK * KK);
  int r = wid % (KK * KK);
  int k = r / KK, j = r % KK;

  const float* tr = T + ((size_t)b * KK + k) * NNODE;
  const float* cr = Cmat + ((size_t)b * KK + j) * NNODE;
  float s = 0.f;
  for (int m = lane; m < NNODE; m += 32) s += tr[m] * cr[m];
#pragma unroll
  for (int off = 16; off > 0; off >>= 1) s += __shfl_down(s, off, 32);
  if (lane == 0) out_adj[(size_t)b * KK * KK + k * KK + j] = fmaxf(s, 0.f);
}

// ---------------------------------------------------------------------------
// Kernel 8: new_node[b,k,o] = sum_d (sum_chunks P)[b,k,d] * lin_w[o,d] + lin_b[o]
// ---------------------------------------------------------------------------
__global__ void k_newnode(const float* __restrict__ Ppart, const float* __restrict__ lin_w,
                          const float* __restrict__ lin_b, float* __restrict__ out_node) {
  int tid = blockIdx.x * blockDim.x + threadIdx.x;
  if (tid >= BB * KK * DD) return;
  int o = tid % DD;
  size_t base = (size_t)(tid / DD) * DD;  // (b*KK + k) * DD
  const size_t cs = (size_t)BB * KK * DD;

  float a = lin_b[o];
  for (int d = 0; d < DD; ++d) {
    float p = Ppart[base + d] + Ppart[cs + base + d] +
              Ppart[2 * cs + base + d] + Ppart[3 * cs + base + d];
    a += p * lin_w[o * DD + d];
  }
  out_node[tid] = a;
}

// ---------------------------------------------------------------------------
extern "C" void kernel_launch(void* const* d_in, const int* in_sizes, int n_in,
                              void* d_out, int out_size, void* d_ws, size_t ws_size,
                              hipStream_t stream) {
  const float* node  = (const float*)d_in[0];
  const float* adj   = (const float*)d_in[1];
  const float* w1    = (const float*)d_in[2];
  const float* b1    = (const float*)d_in[3];
  const float* W2    = (const float*)d_in[4];
  const float* b2    = (const float*)d_in[5];
  const float* W3    = (const float*)d_in[6];
  const float* b3    = (const float*)d_in[7];
  const float* convw = (const float*)d_in[8];
  const float* convb = (const float*)d_in[9];
  const float* linw  = (const float*)d_in[10];
  const float* linb  = (const float*)d_in[11];
  float* out = (float*)d_out;
  float* ws  = (float*)d_ws;

  // workspace layout (floats)
  float* cent  = ws;                   // 16*128*128      = 262144
  float* cn    = cent + 262144;        // 16*128          = 2048
  float* nn_   = cn + 2048;            // 16*4000         = 64000
  float* dotsT = nn_ + 64000;          // 16*4000*128     = 8192000
  float* Cm    = dotsT + 8192000;      // 16*32*4000      = 2048000
  float* T     = Cm + 2048000;         // 16*32*4000      = 2048000
  float* Pp    = T + 2048000;          // 4*16*32*128     = 262144
  // total = 12,878,336 floats ~= 51.5 MB

  k_centroid_mlp<<<(BB * DD + 127) / 128, 128, 0, stream>>>(node, w1, b1, W2, b2, W3, b3, cent);
  k_rownorm128<<<(BB * HK * 32 + 255) / 256, 256, 0, stream>>>(cent, cn, BB * HK);
  k_rownorm128<<<(BB * NNODE * 32 + 255) / 256, 256, 0, stream>>>(node, nn_, BB * NNODE);
  k_dots_wmma<<<(BB * (NNODE / 16) * 32) / 256, 256, 0, stream>>>(node, cent, dotsT);
  k_assign<<<(BB * NNODE + 255) / 256, 256, 0, stream>>>(dotsT, cn, nn_, convw, convb, Cm);
  k_cadj_wmma<<<(BB * (NNODE / 16) * 32) / 256, 256, 0, stream>>>(Cm, adj, T);
  k_cnode_wmma<<<(BB * 8 * 4 * 32) / 256, 256, 0, stream>>>(Cm, node, Pp);
  k_newadj<<<(BB * KK * KK * 32 + 255) / 256, 256, 0, stream>>>(T, Cm, out + BB * KK * DD);
  k_newnode<<<(BB * KK * DD + 255) / 256, 256, 0, stream>>>(Pp, linw, linb, out);
}